// Caps_1486058684775
// MI455X (gfx1250) — compile-verified
//
#include <hip/hip_runtime.h>
#include <math.h>

// Problem constants (from reference)
#define BB 32      // batch
#define II 2048    // input capsules
#define JJ 16      // input capsule dim
#define CC 64      // capsules
#define DD 32      // capsule dim
#define TI 32      // i-tile per workgroup
#define KK (TI*JJ) // 512 = contraction length per tile
#define NIT (II/TI)
#define XS 516     // padded LDS row stride for X  (b-lane reads -> 4-bank stride, conflict free)
#define WTS 40     // padded LDS row stride for W^T (half-wave groups on disjoint banks)

typedef __attribute__((ext_vector_type(2))) float v2f;
typedef __attribute__((ext_vector_type(8))) float v8f;

__device__ __forceinline__ v8f wmma_f32_16x16x4(v2f a, v2f b, v8f c) {
  // D = A(16x4) * B(4x16) + C, fp32 exact path on CDNA5
  return __builtin_amdgcn_wmma_f32_16x16x4_f32(false, a, false, b, (short)0, c, false, false);
}

// --- CDNA5 async global<->LDS DMA (ASYNCcnt-tracked; bypasses VGPRs) -------
__device__ __forceinline__ void async_load_b128(const void* gsrc, void* ldst) {
  // low 32 bits of a flat LDS pointer are the LDS byte address (ISA 10.2)
  const unsigned l = (unsigned)(uintptr_t)ldst;
  asm volatile("global_load_async_to_lds_b128 %0, %1, off"
               :: "v"(l), "v"(gsrc) : "memory");
}
__device__ __forceinline__ void async_store_b32(void* gdst, const void* lsrc) {
  const unsigned l = (unsigned)(uintptr_t)lsrc;
  asm volatile("global_store_async_from_lds_b32 %0, %1, off"
               :: "v"(gdst), "v"(l) : "memory");
}
__device__ __forceinline__ void wait_async0() {
  asm volatile("s_wait_asynccnt 0x0" ::: "memory");
}

// ---------------------------------------------------------------------------
// Pass 1 / Pass 4: S[b,d] += sum_{i in tile, j} A[b,(i,j)] * W[c,i,d,j]
// A = x (pass 1, async-staged) or c2-scaled x (pass 4, manual staging because
// of the in-flight multiply). Accumulate via global f32 atomics.
// ---------------------------------------------------------------------------
template<bool SCALED>
__global__ __launch_bounds__(128)
void caps_gemm_accum(const float* __restrict__ X, const float* __restrict__ W,
                     const float* __restrict__ c2, float* __restrict__ sOut) {
  __shared__ float Xs[BB * XS];     // 66,048 B
  __shared__ float Wt[KK * WTS];    // 81,920 B  (W transposed to [k=(i,j)][d])
  const int i0  = blockIdx.x * TI;
  const int c   = blockIdx.y;
  const int tid = threadIdx.x;

  // Stage X[b, i0:i0+TI, :] -> Xs[b][k], k = i_local*16 + j (contiguous rows)
  for (int t = tid; t < (BB * KK) / 4; t += 128) {
    const int f = t * 4;
    const int b = f >> 9;         // /512
    const int k = f & (KK - 1);
    const float* src = X + (size_t)b * II * JJ + (size_t)i0 * JJ + k;
    if (SCALED) {
      float4 x4 = *(const float4*)src;
      const float cv = c2[(size_t)b * CC * II + (size_t)c * II + i0 + (k >> 4)];
      x4.x *= cv; x4.y *= cv; x4.z *= cv; x4.w *= cv;
      *(float4*)&Xs[b * XS + k] = x4;
    } else {
      async_load_b128(src, &Xs[b * XS + k]);   // DMA straight into LDS
    }
  }
  // Stage W[c, i0:i0+TI, :, :] transposed: Wt[(ii*16+j)*WTS + d] = W[c,i0+ii,d,j]
  const float* Wg = W + (size_t)c * II * DD * JJ + (size_t)i0 * DD * JJ;
  for (int t = tid; t < (TI * DD * JJ) / 4; t += 128) {
    const int f   = t * 4;
    const int ii  = f >> 9;
    const int rem = f & 511;
    const int d   = rem >> 4;
    const int j   = rem & 15;
    const float4 w4 = *(const float4*)(Wg + f);
    const int k = ii * JJ + j;    // j <= 12, so k..k+3 stay in row ii
    Wt[(k + 0) * WTS + d] = w4.x;
    Wt[(k + 1) * WTS + d] = w4.y;
    Wt[(k + 2) * WTS + d] = w4.z;
    Wt[(k + 3) * WTS + d] = w4.w;
  }
  if (!SCALED) wait_async0();     // drain this wave's async LDS writes
  __syncthreads();                // make all waves' staging visible

  const int w = tid >> 5, lane = tid & 31;
  const int lane16 = lane & 15;
  const int lh2 = (lane >> 4) << 1;       // K offset for upper half-wave
  const int lh8 = (lane >> 4) << 3;       // M offset for C/D rows
  const int m0 = (w >> 1) * 16, n0 = (w & 1) * 16;   // wave -> MN quadrant

  v8f acc = {};
  const float* Xr = &Xs[(m0 + lane16) * XS + lh2];
  const float* Wr = &Wt[lh2 * WTS + n0 + lane16];
#pragma unroll 4
  for (int kk = 0; kk < KK; kk += 4) {
    v2f a, b;
    a.x = Xr[kk];
    a.y = Xr[kk + 1];
    b.x = Wr[kk * WTS];
    b.y = Wr[(kk + 1) * WTS];
    acc = wmma_f32_16x16x4(a, b, acc);
  }
#pragma unroll
  for (int r = 0; r < 8; ++r) {
    const int b = m0 + r + lh8;
    atomicAdd(&sOut[(size_t)b * CC * DD + (size_t)c * DD + n0 + lane16], acc[r]);
  }
}

// ---------------------------------------------------------------------------
// Pass 2: logit[b,c,i] = sum_j x[b,i,j] * Q[b,(i,j)],
//         Q[b,(i,j)]   = sum_d v0[b,c,d] * W[c,i,d,j]   (32x512x32 GEMM)
// X and raw-W tiles staged via async DMA; B operand reads the raw W copy
// conflict-free (lanes stride 1 over j). shfl_xor reduces over j.
// ---------------------------------------------------------------------------
__global__ __launch_bounds__(128)
void caps_logits(const float* __restrict__ X, const float* __restrict__ W,
                 const float* __restrict__ v0, float* __restrict__ logits) {
  __shared__ float Xs[BB * XS];        // 66,048 B
  __shared__ float WsR[TI * DD * JJ];  // 65,536 B raw layout [ii][d][j]
  __shared__ float Vs[BB * 33];        // v0[b, c, :] padded
  __shared__ float Ls[BB * 33];        // logit tile [b][i_local]
  const int i0  = blockIdx.x * TI;
  const int c   = blockIdx.y;
  const int tid = threadIdx.x;

  for (int t = tid; t < (BB * KK) / 4; t += 128) {
    const int f = t * 4;
    const int b = f >> 9;
    const int k = f & (KK - 1);
    async_load_b128(X + (size_t)b * II * JJ + (size_t)i0 * JJ + k,
                    &Xs[b * XS + k]);
  }
  const float* Wg = W + (size_t)c * II * DD * JJ + (size_t)i0 * DD * JJ;
  for (int t = tid; t < (TI * DD * JJ) / 4; t += 128) {
    async_load_b128(Wg + t * 4, &WsR[t * 4]);
  }
  for (int t = tid; t < BB * DD; t += 128) {
    const int b = t >> 5, d = t & 31;
    Vs[b * 33 + d] = v0[(size_t)b * CC * DD + (size_t)c * DD + d];
  }
  wait_async0();
  __syncthreads();

  const int w = tid >> 5, lane = tid & 31;
  const int lane16 = lane & 15;
  const int lh2 = (lane >> 4) << 1;
  const int lh8 = (lane >> 4) << 3;

  for (int it = w; it < TI; it += 4) {       // each wave owns 8 i-values
#pragma unroll
    for (int mi = 0; mi < 2; ++mi) {
      const int m0 = mi * 16;
      v8f acc = {};
#pragma unroll
      for (int kk = 0; kk < DD; kk += 4) {   // K = d = 32
        v2f a, b;
        a.x = Vs[(m0 + lane16) * 33 + kk + lh2];
        a.y = Vs[(m0 + lane16) * 33 + kk + 1 + lh2];
        b.x = WsR[it * 512 + (kk + lh2) * 16 + lane16];
        b.y = WsR[it * 512 + (kk + 1 + lh2) * 16 + lane16];
        acc = wmma_f32_16x16x4(a, b, acc);
      }
#pragma unroll
      for (int r = 0; r < 8; ++r) {
        const int b = m0 + r + lh8;
        float v = acc[r] * Xs[b * XS + it * JJ + lane16];  // * x[b, i, j=lane]
        v += __shfl_xor(v, 1);
        v += __shfl_xor(v, 2);
        v += __shfl_xor(v, 4);
        v += __shfl_xor(v, 8);               // sum over 16 j-lanes
        if (lane16 == 0) Ls[b * 33 + it] = v;
      }
    }
  }
  __syncthreads();
  // async LDS -> global write-out of the logit tile
  for (int t = tid; t < BB * TI; t += 128) {
    const int b = t >> 5, il = t & 31;
    async_store_b32(logits + (size_t)b * CC * II + (size_t)c * II + i0 + il,
                    &Ls[b * 33 + il]);
  }
  wait_async0();   // (S_ENDPGM would wait-idle anyway)
}

// ---------------------------------------------------------------------------
// Pass 3: softmax over the capsule axis (C=64) for each (b, i), in place.
// ---------------------------------------------------------------------------
__global__ __launch_bounds__(256)
void caps_softmax(float* __restrict__ logits) {
  const int g = blockIdx.x * 256 + threadIdx.x;   // over B*I = 65536
  const int b = g >> 11;
  const int i = g & 2047;
  float* p = logits + (size_t)b * CC * II + i;
  float m = -3.4e38f;
#pragma unroll
  for (int cc = 0; cc < CC; ++cc) m = fmaxf(m, p[(size_t)cc * II]);
  float s = 0.f;
#pragma unroll
  for (int cc = 0; cc < CC; ++cc) s += expf(p[(size_t)cc * II] - m);
  const float inv = 1.f / s;
#pragma unroll
  for (int cc = 0; cc < CC; ++cc) {
    const float e = expf(p[(size_t)cc * II] - m);
    p[(size_t)cc * II] = e * inv;
  }
}

// ---------------------------------------------------------------------------
// squash(s * cscale) -> v    (cscale = 1/64 for round 0, 1 for final)
// ---------------------------------------------------------------------------
__global__ __launch_bounds__(256)
void caps_squash(const float* __restrict__ s, float* __restrict__ v, float cscale) {
  const int g = blockIdx.x * 256 + threadIdx.x;   // over B*C = 2048
  const float* sp = s + (size_t)g * DD;
  float* vp = v + (size_t)g * DD;
  float t[DD];
  float sn = 0.f;
#pragma unroll
  for (int d = 0; d < DD; ++d) { t[d] = sp[d] * cscale; sn += t[d] * t[d]; }
  const float sc = sn / ((1.f + sn) * sqrtf(sn + 1e-7f));
#pragma unroll
  for (int d = 0; d < DD; ++d) vp[d] = t[d] * sc;
}

// ---------------------------------------------------------------------------
extern "C" void kernel_launch(void* const* d_in, const int* in_sizes, int n_in,
                              void* d_out, int out_size, void* d_ws, size_t ws_size,
                              hipStream_t stream) {
  const float* X = (const float*)d_in[0];   // [32, 2048, 16]
  const float* W = (const float*)d_in[1];   // [64, 2048, 32, 16]
  float* out = (float*)d_out;               // [32, 64, 32]

  float* s0 = (float*)d_ws;                 // 65536 f32
  float* v0 = s0 + (size_t)BB * CC * DD;    // 65536 f32
  float* s2 = v0 + (size_t)BB * CC * DD;    // 65536 f32
  float* lg = s2 + (size_t)BB * CC * DD;    // 4,194,304 f32 (16 MB)

  hipMemsetAsync(s0, 0, (size_t)BB * CC * DD * sizeof(float), stream);
  hipMemsetAsync(s2, 0, (size_t)BB * CC * DD * sizeof(float), stream);

  const dim3 gG(NIT, CC), bG(128);
  caps_gemm_accum<false><<<gG, bG, 0, stream>>>(X, W, nullptr, s0);
  caps_squash<<<dim3((BB * CC) / 256), dim3(256), 0, stream>>>(s0, v0, 1.0f / CC);
  caps_logits<<<gG, bG, 0, stream>>>(X, W, v0, lg);
  caps_softmax<<<dim3((BB * II) / 256), dim3(256), 0, stream>>>(lg);
  caps_gemm_accum<true><<<gG, bG, 0, stream>>>(X, W, lg, s2);
  caps_squash<<<dim3((BB * CC) / 256), dim3(256), 0, stream>>>(s2, out, 1.0f);
}